// MyRelConv_77214922048100
// MI455X (gfx1250) — compile-verified
//
#include <hip/hip_runtime.h>
#include <math.h>

// ---------------------------------------------------------------------------
// MyRelConv (GAT-style relational conv) for MI455X / gfx1250, wave32.
//
// Pipeline (all on `stream`, graph-capture safe, no host-side sync):
//   K0 init      : m=-inf, denom=0, deg=0, agg=0
//   K1 node_attn : an = nt @ attn_n_param                    (N x 9)
//   K2 edge_logit: e = leakyrelu(dot(an[src], nt[dst]));  segment-max via
//                  float-bits atomic max/min                 (E)
//   K3 edge_exp  : ex = exp(e - m[dst]); denom += ex; deg += 1 (atomics)
//   K4 scatter   : agg[dst] += feat[src] * (ex/denom[dst])   (wave per edge)
//   K5 norm      : h_neigh = agg / max(deg,1)   (in place)
//   K6 WMMA GEMM : out = feat@W_self^T + h_neigh@W_neigh^T + b_self + b_neigh
//                  using V_WMMA_F32_16X16X4_F32 (exact f32 — GEMM is only
//                  ~3.3 GFLOP, the op is L2/atomic bound, so full precision
//                  costs nothing).
// ---------------------------------------------------------------------------

#define N_NODES   50000
#define N_EDGES   800000
#define IN_FEATS  128
#define OUT_FEATS 128
#define NTY       9
#define NEG_SLOPE 0.2f

typedef float v2f __attribute__((ext_vector_type(2)));
typedef float v8f __attribute__((ext_vector_type(8)));

// ---------------- K0: init scratch ----------------
__global__ void k0_init(int* __restrict__ mbits, float* __restrict__ denom,
                        float* __restrict__ deg, float* __restrict__ agg) {
    int i = blockIdx.x * blockDim.x + threadIdx.x;
    if (i < N_NODES) {
        mbits[i] = 0xFF800000;     // -inf bits
        denom[i] = 0.0f;
        deg[i]   = 0.0f;
    }
    if (i < N_NODES * IN_FEATS) agg[i] = 0.0f;
}

// ---------------- K1: an = nt @ attn (N x 9) ----------------
__global__ void k1_node_attn(const float* __restrict__ nt,
                             const float* __restrict__ attn,
                             float* __restrict__ an) {
    int i = blockIdx.x * blockDim.x + threadIdx.x;
    if (i >= N_NODES) return;
    float v[NTY];
#pragma unroll
    for (int k = 0; k < NTY; ++k) v[k] = nt[i * NTY + k];
#pragma unroll
    for (int j = 0; j < NTY; ++j) {
        float s = 0.0f;
#pragma unroll
        for (int k = 0; k < NTY; ++k) s += v[k] * attn[k * NTY + j];
        an[i * NTY + j] = s;
    }
}

// ---------------- K2: edge logits + segment max ----------------
__global__ void k2_edge_logit(const float* __restrict__ an,
                              const float* __restrict__ nt,
                              const int* __restrict__ src,
                              const int* __restrict__ dst,
                              float* __restrict__ e,
                              int* __restrict__ mbits) {
    int i = blockIdx.x * blockDim.x + threadIdx.x;
    if (i >= N_EDGES) return;
    int s = src[i], d = dst[i];
    float acc = 0.0f;
#pragma unroll
    for (int j = 0; j < NTY; ++j) acc += an[s * NTY + j] * nt[d * NTY + j];
    float ee = (acc >= 0.0f) ? acc : NEG_SLOPE * acc;
    e[i] = ee;
    // IEEE float atomic-max via integer atomics (no CAS loop):
    // non-negative floats order like signed ints; negative floats order
    // inversely like unsigned ints.
    if (ee >= 0.0f) atomicMax(&mbits[d], __float_as_int(ee));
    else            atomicMin((unsigned int*)&mbits[d], __float_as_uint(ee));
}

// ---------------- K3: ex = exp(e - m[dst]); denom, deg ----------------
__global__ void k3_edge_exp(const int* __restrict__ dst,
                            const int* __restrict__ mbits,
                            float* __restrict__ e,          // in: e, out: ex
                            float* __restrict__ denom,
                            float* __restrict__ deg) {
    int i = blockIdx.x * blockDim.x + threadIdx.x;
    if (i >= N_EDGES) return;
    int d  = dst[i];
    int mb = mbits[d];
    // where(isfinite(m), m, 0): exponent all-ones => inf/nan => 0
    float m = ((mb & 0x7F800000) == 0x7F800000) ? 0.0f : __int_as_float(mb);
    float x = expf(e[i] - m);
    e[i] = x;
    atomicAdd(&denom[d], x);
    atomicAdd(&deg[d], 1.0f);
}

// ---------------- K4: agg[dst] += feat[src] * a  (one wave32 per edge) ----
__global__ void k4_scatter(const float* __restrict__ feat,
                           const int* __restrict__ src,
                           const int* __restrict__ dst,
                           const float* __restrict__ ex,
                           const float* __restrict__ denom,
                           float* __restrict__ agg) {
    int gid  = blockIdx.x * blockDim.x + threadIdx.x;
    int edge = gid >> 5;          // wave-uniform -> scalar loads
    int lane = gid & 31;
    if (edge >= N_EDGES) return;
    int s = src[edge], d = dst[edge];
    float a = ex[edge] / denom[d];
    const float4 f = ((const float4*)(feat + (size_t)s * IN_FEATS))[lane];
    float* o = agg + (size_t)d * IN_FEATS + lane * 4;
    atomicAdd(o + 0, f.x * a);
    atomicAdd(o + 1, f.y * a);
    atomicAdd(o + 2, f.z * a);
    atomicAdd(o + 3, f.w * a);
}

// ---------------- K5: h_neigh = agg / max(deg,1)  (in place) ----------------
__global__ void k5_norm(float* __restrict__ agg, const float* __restrict__ deg) {
    int i = blockIdx.x * blockDim.x + threadIdx.x;
    if (i >= N_NODES * IN_FEATS) return;
    int n = i / IN_FEATS;
    agg[i] = agg[i] / fmaxf(deg[n], 1.0f);
}

// ---------------- K6: fused dual GEMM with f32 WMMA ----------------
// One wave per 16x16 output tile, computed TRANSPOSED:
//   D[o_local][m_local] = sum_k W[o][k] * X[m][k]
// A fragment = rows of W (contiguous), B fragment = rows of X (contiguous).
// f32 16x16x4 striping: lane<16 -> idx-th row, K = {0,1}; lane>=16 -> K = {2,3}.
// D: VGPR g, lane L -> M = g + 8*(L>>4), N = L&15  => each lane stores 8
// consecutive floats of one output row.
__global__ void k6_gemm_wmma(const float* __restrict__ feat,
                             const float* __restrict__ hneigh,
                             const float* __restrict__ Wself,
                             const float* __restrict__ Wneigh,
                             const float* __restrict__ bself,
                             const float* __restrict__ bneigh,
                             float* __restrict__ out) {
    const int wave = threadIdx.x >> 5;
    const int lane = threadIdx.x & 31;
    const int tile = blockIdx.x * 8 + wave;           // 8 waves per block
    const int OUT_TILES = OUT_FEATS / 16;             // 8
    const int node_tile = tile / OUT_TILES;           // 0..3124 (exact)
    const int out_tile  = tile % OUT_TILES;

    const int half = lane >> 4;                       // 0 or 1
    const int idx  = lane & 15;

    const int node0 = node_tile * 16;
    const int o0    = out_tile * 16;

    const float* a_self = Wself  + (size_t)(o0 + idx)   * IN_FEATS;
    const float* a_nei  = Wneigh + (size_t)(o0 + idx)   * IN_FEATS;
    const float* b_feat = feat   + (size_t)(node0 + idx) * IN_FEATS;
    const float* b_hn   = hneigh + (size_t)(node0 + idx) * IN_FEATS;

    v8f acc = {0.f, 0.f, 0.f, 0.f, 0.f, 0.f, 0.f, 0.f};

#pragma unroll 4
    for (int k = 0; k < IN_FEATS; k += 4) {
        const int kk = k + 2 * half;                  // per-lane K pair
        v2f aS = *(const v2f*)(a_self + kk);
        v2f bF = *(const v2f*)(b_feat + kk);
        v2f aN = *(const v2f*)(a_nei + kk);
        v2f bH = *(const v2f*)(b_hn + kk);
        // D = A x B + C ; chain both GEMMs into one accumulator.
        acc = __builtin_amdgcn_wmma_f32_16x16x4_f32(
                  false, aS, false, bF, (short)0, acc, false, false);
        acc = __builtin_amdgcn_wmma_f32_16x16x4_f32(
                  false, aN, false, bH, (short)0, acc, false, false);
    }

    // Lane writes out[node0+idx][o0 + 8*half + g], g = 0..7 (contiguous).
    const int obase = o0 + 8 * half;
    float* orow = out + (size_t)(node0 + idx) * OUT_FEATS + obase;
#pragma unroll
    for (int g = 0; g < 8; ++g) {
        orow[g] = acc[g] + bself[obase + g] + bneigh[obase + g];
    }
}

// ---------------------------------------------------------------------------
extern "C" void kernel_launch(void* const* d_in, const int* in_sizes, int n_in,
                              void* d_out, int out_size, void* d_ws, size_t ws_size,
                              hipStream_t stream) {
    const float* feat   = (const float*)d_in[0];   // N x 128
    const float* ntype  = (const float*)d_in[1];   // N x 1 x 9 (== N x 9)
    const int*   src    = (const int*)  d_in[2];   // E
    const int*   dst    = (const int*)  d_in[3];   // E
    const float* attn   = (const float*)d_in[4];   // 9 x 9
    const float* Wself  = (const float*)d_in[5];   // 128 x 128
    const float* bself  = (const float*)d_in[6];   // 128
    const float* Wneigh = (const float*)d_in[7];   // 128 x 128
    const float* bneigh = (const float*)d_in[8];   // 128
    float*       out    = (float*)d_out;           // N x 128

    // Workspace layout (all offsets 16B aligned).
    float* ws    = (float*)d_ws;
    float* an    = ws;                              // N*9   = 450000
    float* e     = an + (size_t)N_NODES * NTY;      // E     = 800000 (e, then ex)
    int*   mbits = (int*)(e + N_EDGES);             // N
    float* denom = (float*)(mbits + N_NODES);       // N
    float* deg   = denom + N_NODES;                 // N
    float* agg   = deg + N_NODES;                   // N*128 (then h_neigh in place)

    const int T = 256;
    const int gInit   = (N_NODES * IN_FEATS + T - 1) / T;   // 25000
    const int gNode   = (N_NODES + T - 1) / T;              // 196
    const int gEdge   = (N_EDGES + T - 1) / T;              // 3125
    const int gScat   = (N_EDGES * 32 + T - 1) / T;         // 100000
    const int gNorm   = gInit;
    const int gGemm   = (N_NODES / 16) * (OUT_FEATS / 16) / 8; // 3125 (exact)

    k0_init      <<<gInit, T, 0, stream>>>(mbits, denom, deg, agg);
    k1_node_attn <<<gNode, T, 0, stream>>>(ntype, attn, an);
    k2_edge_logit<<<gEdge, T, 0, stream>>>(an, ntype, src, dst, e, mbits);
    k3_edge_exp  <<<gEdge, T, 0, stream>>>(dst, mbits, e, denom, deg);
    k4_scatter   <<<gScat, T, 0, stream>>>(feat, src, dst, e, denom, agg);
    k5_norm      <<<gNorm, T, 0, stream>>>(agg, deg);
    k6_gemm_wmma <<<gGemm, T, 0, stream>>>(feat, agg, Wself, Wneigh,
                                           bself, bneigh, out);
}